// VoronoiBaseNN_81286551044462
// MI455X (gfx1250) — compile-verified
//
#include <hip/hip_runtime.h>

#define KNN 11
#define NPTS 50000
#define NSP  4000
#define CHUNK_SP 400           // spoints per LDS chunk = 25 tiles of 16
#define CHUNK_TILES 25
#define NCHUNKS 10             // 4000 / 400
#define WAVES 8
#define BLOCK (WAVES * 32)
#define PTS_PER_BLOCK (WAVES * 16)

typedef __attribute__((ext_vector_type(2))) float v2f;
typedef __attribute__((ext_vector_type(8))) float v8f;
typedef __attribute__((ext_vector_type(4))) int   v4i;

#define AS_GLOBAL __attribute__((address_space(1)))
#define AS_LOCAL  __attribute__((address_space(3)))

// Precompute (x, y, z, x^2+y^2+z^2) per spoint so the WMMA K=4 slot carries ||s||^2.
__global__ void build_s4(const float* __restrict__ sp, float4* __restrict__ s4, int n) {
    int i = blockIdx.x * blockDim.x + threadIdx.x;
    if (i < n) {
        float x = sp[i * 3 + 0], y = sp[i * 3 + 1], z = sp[i * 3 + 2];
        s4[i] = make_float4(x, y, z, x * x + y * y + z * z);
    }
}

__global__ __launch_bounds__(BLOCK) void knn_voronoi(
    const float* __restrict__ points,    // (2, NPTS, 3)
    const float4* __restrict__ s4g,      // (2, NSP) : x,y,z,|s|^2
    float* __restrict__ out)             // (2, NPTS)
{
    __shared__ float4 s_chunk[CHUNK_SP];            // 6.4 KB
    __shared__ float  s_md[WAVES][32][KNN];         // merge scratch
    __shared__ int    s_mi[WAVES][32][KNN];

    const int batch = blockIdx.y;
    const int lane  = threadIdx.x & 31;
    const int wave  = threadIdx.x >> 5;
    const bool hi   = lane >= 16;
    const int  l16  = lane & 15;

    const float4* S4 = s4g + batch * NSP;
    const float*  P  = points + (size_t)batch * NPTS * 3;

    // This lane's point (column N = l16 of the wave's 16-point tile).
    const int ptile  = blockIdx.x * PTS_PER_BLOCK + wave * 16;
    const int pidx   = ptile + l16;
    const int pclamp = pidx < NPTS ? pidx : NPTS - 1;
    const float px = P[pclamp * 3 + 0];
    const float py = P[pclamp * 3 + 1];
    const float pz = P[pclamp * 3 + 2];

    // B fragment (4x16): rows K={0,1} in lanes 0-15, K={2,3} in lanes 16-31.
    // B col = (-2px, -2py, -2pz, 1)  =>  D = ||s||^2 - 2 p.s  (rank-equivalent to d2)
    v2f bfrag;
    bfrag.x = hi ? (-2.0f * pz) : (-2.0f * px);
    bfrag.y = hi ? 1.0f         : (-2.0f * py);

    float dk[KNN];
    int   ik[KNN];
#pragma unroll
    for (int k = 0; k < KNN; ++k) { dk[k] = 3.0e38f; ik[k] = 0; }

    const int afoff = l16 * 4 + (hi ? 2 : 0);   // A-fragment float offset within a tile

    for (int c = 0; c < NCHUNKS; ++c) {
        // Stage 400 spoints (6.4 KB) into LDS, shared by all 8 waves.
        for (int i = threadIdx.x; i < CHUNK_SP; i += BLOCK) {
#if __has_builtin(__builtin_amdgcn_global_load_async_to_lds_b128)
            __builtin_amdgcn_global_load_async_to_lds_b128(
                (AS_GLOBAL v4i*)(S4 + c * CHUNK_SP + i),
                (AS_LOCAL v4i*)&s_chunk[i], 0, 0);
#else
            s_chunk[i] = S4[c * CHUNK_SP + i];
#endif
        }
        if (c + 1 < NCHUNKS)
            __builtin_prefetch(S4 + (c + 1) * CHUNK_SP, 0, 1);
#if __has_builtin(__builtin_amdgcn_s_wait_asynccnt)
        __builtin_amdgcn_s_wait_asynccnt(0);
#else
        asm volatile("s_wait_asynccnt 0" ::: "memory");
#endif
        __syncthreads();

        const float* lsp = (const float*)s_chunk;
        // Software-pipelined A-fragment: load tile t+1 while processing tile t.
        v2f afrag = *(const v2f*)&lsp[afoff];
#pragma unroll 1
        for (int t = 0; t < CHUNK_TILES; ++t) {
            v8f acc = {};
            acc = __builtin_amdgcn_wmma_f32_16x16x4_f32(
                false, afrag, false, bfrag, (short)0, acc, false, false);
            if (t + 1 < CHUNK_TILES)
                afrag = *(const v2f*)&lsp[(t + 1) * 64 + afoff];

            // Group qualification: one compare+branch on the tree-min of 8 outputs.
            float gmin = fminf(fminf(fminf(acc[0], acc[1]), fminf(acc[2], acc[3])),
                               fminf(fminf(acc[4], acc[5]), fminf(acc[6], acc[7])));
            if (gmin < dk[KNN - 1]) {
                const int base = c * CHUNK_SP + t * 16 + (hi ? 8 : 0);
#pragma unroll
                for (int v = 0; v < 8; ++v) {
                    float d = acc[v];
                    if (d < dk[KNN - 1]) {
                        dk[KNN - 1] = d;
                        ik[KNN - 1] = base + v;
#pragma unroll
                        for (int k = KNN - 1; k > 0; --k) {
                            if (dk[k] < dk[k - 1]) {
                                float td = dk[k]; dk[k] = dk[k - 1]; dk[k - 1] = td;
                                int   ti = ik[k]; ik[k] = ik[k - 1]; ik[k - 1] = ti;
                            }
                        }
                    }
                }
            }
        }
        __syncthreads();
    }

    // Dump per-lane top-11 (each lane saw half the spoints for its point).
#pragma unroll
    for (int k = 0; k < KNN; ++k) {
        s_md[wave][lane][k] = dk[k];
        s_mi[wave][lane][k] = ik[k];
    }
    __syncthreads();

    if (!hi) {
        // Merge the two sorted 11-lists for this point.
        const float* da = s_md[wave][lane];
        const int*   ia = s_mi[wave][lane];
        const float* db = s_md[wave][lane + 16];
        const int*   ib = s_mi[wave][lane + 16];
        int fi[KNN];
        int a = 0, bb = 0;
#pragma unroll
        for (int k = 0; k < KNN; ++k) {
            float xa = da[a], xb = db[bb];
            if (xa <= xb) { fi[k] = ia[a]; ++a; }
            else          { fi[k] = ib[bb]; ++bb; }
        }

        // Voronoi-edge epilogue (spoint coords from the f32 table; exact f32 math).
        float4 s0 = S4[fi[0]];
        float cx = px - s0.x, cy = py - s0.y, cz = pz - s0.z;
        float best = 3.0e38f;
#pragma unroll
        for (int j = 1; j < KNN; ++j) {
            float4 sj = S4[fi[j]];
            float ex = sj.x - s0.x, ey = sj.y - s0.y, ez = sj.z - s0.z;
            float el2 = ex * ex + ey * ey + ez * ez;
            float el  = sqrtf(el2);
            float vl  = (cx * ex + cy * ey + cz * ez) / el;
            float tt  = vl - 0.5f * el;
            best = fminf(best, tt * tt);
        }
        if (pidx < NPTS) out[batch * NPTS + pidx] = best;
    }
}

extern "C" void kernel_launch(void* const* d_in, const int* in_sizes, int n_in,
                              void* d_out, int out_size, void* d_ws, size_t ws_size,
                              hipStream_t stream) {
    const float* points  = (const float*)d_in[0];   // (2, 50000, 3)
    const float* spoints = (const float*)d_in[1];   // (2, 4000, 3)
    float* out = (float*)d_out;                     // (2, 50000)
    float4* s4 = (float4*)d_ws;                     // 2*4000 float4 = 128 KB scratch

    const int nsp_total = 2 * NSP;
    build_s4<<<(nsp_total + 255) / 256, 256, 0, stream>>>(spoints, s4, nsp_total);

    dim3 grid((NPTS + PTS_PER_BLOCK - 1) / PTS_PER_BLOCK, 2);
    knn_voronoi<<<grid, BLOCK, 0, stream>>>(points, s4, out);
}